// GlobalAttentionBlock_83399674954094
// MI455X (gfx1250) — compile-verified
//
#include <hip/hip_runtime.h>
#include <hip/hip_bf16.h>

// ---- problem constants -------------------------------------------------
#define BATCH 8
#define DMODEL 1024
#define HW 4096          // 64*64
#define SLEN 4097        // HW + 1
#define NHEAD 16
#define HDIM 64
#define SCALE 0.125f     // 1/sqrt(64)
#define SST 4104         // padded stride for scores rows

typedef float v2f __attribute__((ext_vector_type(2)));
typedef float v8f __attribute__((ext_vector_type(8)));

// ---- workspace layout (floats) ----------------------------------------
#define OFF_PE    0u           // peT[d][pos]            4,194,304
#define OFF_QKV0  4194304u     // qkv0[b][3072]             24,576
#define OFF_U     4218880u     // u[b][h][d]               131,072
#define OFF_QBK   4349952u     // q0.bk  [b][h]                128
#define OFF_SC    4350080u     // scores[b][h][SST]        525,312
#define OFF_MP    4875392u     // m_part[pc8][b][h][d]   1,048,576
#define OFF_CA    5923968u     // cls_attn[b][1024]          8,192
#define OFF_TR    5932160u     // tok_row [b][1024]          8,192

// =======================================================================
// Kernel 1: positional encoding, stored TRANSPOSED (d-major, pos inner)
// so it has the same layout as x -> coalesced streaming adds.
// =======================================================================
__global__ void pe_kernel(float* __restrict__ peT) {
  unsigned i = blockIdx.x * 256u + threadIdx.x;   // 0 .. D*HW-1
  int d   = (int)(i >> 12);
  int pos = (int)(i & 4095u);
  // rate = 10000^(-(2*(d/2))/1024);  ln(10000)=9.210340371976184
  float rate = __expf((float)(d >> 1) * (-2.0f * 9.210340371976184f / 1024.0f));
  float ang  = (float)pos * rate;
  peT[i] = (d & 1) ? __cosf(ang) : __sinf(ang);
}

// =======================================================================
// Kernel 2: qkv0 = ctx @ Wqkv^T + bqkv   (8x1024 x 1024x3072)
// WMMA f32 16x16x4, M=16 (8 real batches + 8 zero rows), one wave/tile.
// Branch-free A-row masking: rows 8..15 read a clamped (valid) address
// and are multiplied by 0 -> no EXEC divergence around the WMMA loop.
// =======================================================================
__global__ void qkv0_wmma_kernel(const float* __restrict__ ctx,
                                 const float* __restrict__ Wqkv,
                                 const float* __restrict__ bqkv,
                                 float* __restrict__ qkv0) {
  const int n0    = blockIdx.x * 16;
  const int lane  = threadIdx.x & 31;
  const int mrow  = lane & 15;              // A row / B col
  const int khalf = (lane >> 4) << 1;       // 0 or 2
  const float msk = (mrow < BATCH) ? 1.0f : 0.0f;
  const float* arow = ctx + (mrow & 7) * DMODEL;   // always valid
  v8f acc = {};
  for (int kk = 0; kk < DMODEL; kk += 4) {
    const int k0 = kk + khalf;
    v2f a, b;
    a.x = arow[k0]     * msk;
    a.y = arow[k0 + 1] * msk;
    const float* wr = Wqkv + (size_t)(n0 + mrow) * DMODEL + k0;
    b.x = wr[0];
    b.y = wr[1];
    acc = __builtin_amdgcn_wmma_f32_16x16x4_f32(false, a, false, b,
                                                (short)0, acc, false, false);
  }
  const int col   = lane & 15;
  const int hbase = (lane >> 4) * 8;
#pragma unroll
  for (int v = 0; v < 8; ++v) {
    int mo = v + hbase;
    if (mo < BATCH)
      qkv0[mo * 3072 + n0 + col] = acc[v] + bqkv[n0 + col];
  }
}

// =======================================================================
// Kernel 3: u[b,h,:] = q0[b,h] @ Wk_h ; qbk[b,h] = q0[b,h].bk_h
// =======================================================================
__global__ void u_kernel(const float* __restrict__ qkv0,
                         const float* __restrict__ Wqkv,
                         const float* __restrict__ bqkv,
                         float* __restrict__ u,
                         float* __restrict__ qbk) {
  const int bh = blockIdx.x, b = bh >> 4, h = bh & 15;
  const int t  = threadIdx.x;
  __shared__ float q0s[64];
  __shared__ float red[64];
  if (t < 64) q0s[t] = qkv0[b * 3072 + h * HDIM + t];
  __syncthreads();
  if (t < 64) red[t] = q0s[t] * bqkv[DMODEL + h * HDIM + t];
  __syncthreads();
  if (t == 0) {
    float s = 0.f;
    for (int j = 0; j < 64; ++j) s += red[j];
    qbk[bh] = s;
  }
  for (int d = t; d < DMODEL; d += 256) {
    float s = 0.f;
    for (int j = 0; j < 64; ++j)
      s += q0s[j] * Wqkv[(size_t)(DMODEL + h * HDIM + j) * DMODEL + d];
    u[(size_t)bh * DMODEL + d] = s;
  }
}

// =======================================================================
// Kernel 4: streaming scores pass (the big one): for pos=0..4095
//   scores[b,h,1+pos] = SCALE*(u[b,h,:].(x[b,:,pos]+peT[:,pos]) + qbk)
// GEMM per wave: D(16h x 16pos) += A(u,16x4) x B(xc,4x16). K=1024.
// u is staged transposed in LDS; xc streamed coalesced from HBM.
// =======================================================================
__global__ void scores_wmma_kernel(const float* __restrict__ x,
                                   const float* __restrict__ peT,
                                   const float* __restrict__ u,
                                   const float* __restrict__ qbk,
                                   float* __restrict__ scores) {
  const int b     = blockIdx.y;
  const int chunk = blockIdx.x;            // 32 chunks * 128 pos
  const int t     = threadIdx.x;           // 256 = 8 waves
  const int wave  = t >> 5;
  const int lane  = t & 31;
  extern __shared__ float uT[];            // [1024][16]  (k-major)
  const float* ub = u + (size_t)b * NHEAD * DMODEL;
  for (int i = 0; i < 64; ++i) {
    int e = i * 256 + t;                   // e = k*16 + h
    int h = e & 15, k = e >> 4;
    uT[e] = ub[h * DMODEL + k];
  }
  __syncthreads();

  const int pos0  = chunk * 128 + wave * 16;
  const int mrow  = lane & 15;
  const int khalf = (lane >> 4) << 1;
  const float* xb = x + (size_t)b * DMODEL * HW;
  v8f acc = {};
  for (int kk = 0; kk < DMODEL; kk += 4) {
    const int k0 = kk + khalf;
    v2f a, bb;
    a.x = uT[k0 * 16 + mrow];
    a.y = uT[(k0 + 1) * 16 + mrow];
    const float* xp = xb  + (size_t)k0 * HW + pos0 + mrow;
    const float* pp = peT + (size_t)k0 * HW + pos0 + mrow;
    bb.x = xp[0]  + pp[0];
    bb.y = xp[HW] + pp[HW];
    if ((kk & 15) == 0)                       // gfx1250 global_prefetch_b8
      __builtin_prefetch(xp + 16 * HW, 0, 1);
    acc = __builtin_amdgcn_wmma_f32_16x16x4_f32(false, a, false, bb,
                                                (short)0, acc, false, false);
  }
  const int col   = pos0 + (lane & 15);
  const int hbase = (lane >> 4) * 8;
#pragma unroll
  for (int v = 0; v < 8; ++v) {
    int h = v + hbase;
    scores[(size_t)(b * NHEAD + h) * SST + 1 + col] =
        SCALE * (acc[v] + qbk[b * NHEAD + h]);
  }
}

// =======================================================================
// Kernel 5: per-(b,h) softmax over 4097 scores; also fills s=0 (q0.k0).
// =======================================================================
__global__ void softmax_kernel(const float* __restrict__ qkv0,
                               float* __restrict__ scores) {
  const int bh = blockIdx.x, b = bh >> 4, h = bh & 15;
  const int t  = threadIdx.x;              // 256
  __shared__ float red[256];
  float* row = scores + (size_t)bh * SST;
  float p = 0.f;
  if (t < 64) {
    float q = qkv0[b * 3072 + h * HDIM + t];
    float k = qkv0[b * 3072 + DMODEL + h * HDIM + t];
    p = q * k;
  }
  red[t] = p; __syncthreads();
  for (int o = 128; o > 0; o >>= 1) { if (t < o) red[t] += red[t + o]; __syncthreads(); }
  if (t == 0) row[0] = SCALE * red[0];
  __syncthreads();

  float mx = -1e30f;
  for (int s = t; s < SLEN; s += 256) mx = fmaxf(mx, row[s]);
  red[t] = mx; __syncthreads();
  for (int o = 128; o > 0; o >>= 1) { if (t < o) red[t] = fmaxf(red[t], red[t + o]); __syncthreads(); }
  mx = red[0]; __syncthreads();

  float sum = 0.f;
  for (int s = t; s < SLEN; s += 256) {
    float e = __expf(row[s] - mx);
    row[s] = e; sum += e;
  }
  red[t] = sum; __syncthreads();
  for (int o = 128; o > 0; o >>= 1) { if (t < o) red[t] += red[t + o]; __syncthreads(); }
  float inv = 1.0f / red[0];
  for (int s = t; s < SLEN; s += 256) row[s] *= inv;
}

// =======================================================================
// Kernel 6: m_part[pc][b][h][d] = sum over pos-chunk of probs*xc
// LDS-staged coalesced x tiles; deterministic partial buffers (no atomics)
// =======================================================================
__global__ void maccum_kernel(const float* __restrict__ x,
                              const float* __restrict__ peT,
                              const float* __restrict__ probs,
                              float* __restrict__ m_part) {
  const int b = blockIdx.z, dc = blockIdx.y, pc = blockIdx.x;
  const int d0 = dc * 128, p0 = pc * 512;
  const int t = threadIdx.x;
  extern __shared__ float lds[];
  float* prs = lds;                    // [16][513]
  float* xcs = lds + 16 * 513;         // [128][65]
  for (int i = 0; i < 32; ++i) {
    int e = i * 256 + t;               // 8192 = 16*512
    int h = e >> 9, p = e & 511;
    prs[h * 513 + p] = probs[(size_t)(b * NHEAD + h) * SST + 1 + p0 + p];
  }
  const int h = t & 15, dg = t >> 4;
  float acc[8] = {0.f,0.f,0.f,0.f,0.f,0.f,0.f,0.f};
  for (int ps = 0; ps < 8; ++ps) {
    __syncthreads();
    for (int i = 0; i < 32; ++i) {
      int e = i * 256 + t;             // 8192 = 128*64
      int pl = e & 63, dl = e >> 6;
      size_t gi = ((size_t)(b * DMODEL + d0 + dl)) * HW + p0 + ps * 64 + pl;
      xcs[dl * 65 + pl] = x[gi] + peT[gi - (size_t)b * DMODEL * HW];
    }
    __syncthreads();
    for (int p = 0; p < 64; ++p) {
      float pr = prs[h * 513 + ps * 64 + p];
#pragma unroll
      for (int j = 0; j < 8; ++j)
        acc[j] += pr * xcs[(dg * 8 + j) * 65 + p];
    }
  }
  float* mr = m_part + ((size_t)(pc * 128) + b * NHEAD + h) * DMODEL + d0 + dg * 8;
#pragma unroll
  for (int j = 0; j < 8; ++j) mr[j] = acc[j];
}

// =======================================================================
// Kernel 7: finish m (add s=0 term), cls_attn = Wv.m + bv   (per batch)
// =======================================================================
__global__ void clsA_kernel(const float* __restrict__ m_part,
                            const float* __restrict__ probs,
                            const float* __restrict__ ctx,
                            const float* __restrict__ Wqkv,
                            const float* __restrict__ bqkv,
                            float* __restrict__ cls_attn) {
  const int b = blockIdx.x, t = threadIdx.x;
  extern __shared__ float mls[];       // [16][1024]
  __shared__ float p0s[16];
  if (t < 16) p0s[t] = probs[(size_t)(b * NHEAD + t) * SST];
  __syncthreads();
  for (int i = 0; i < 64; ++i) {
    int e = i * 256 + t;               // 16384
    int h = e >> 10, d = e & 1023;
    float s = p0s[h] * ctx[b * DMODEL + d];
    for (int pc = 0; pc < 8; ++pc)
      s += m_part[((size_t)(pc * 128) + b * NHEAD + h) * DMODEL + d];
    mls[e] = s;
  }
  __syncthreads();
  for (int j = t; j < DMODEL; j += 256) {
    int h = j >> 6;
    float s = bqkv[2048 + j];
    const float* wr = Wqkv + (size_t)(2048 + j) * DMODEL;
    for (int d = 0; d < DMODEL; ++d) s += wr[d] * mls[h * DMODEL + d];
    cls_attn[b * DMODEL + j] = s;
  }
}

// =======================================================================
// Kernel 8: project through Wo: tok_row = Wo.v_cls + bo ;
// cls row: residual + LayerNorm -> context_token_out
// =======================================================================
__global__ void clsB_kernel(const float* __restrict__ cls_attn,
                            const float* __restrict__ qkv0,
                            const float* __restrict__ ctx,
                            const float* __restrict__ Wo,
                            const float* __restrict__ bo,
                            const float* __restrict__ gamma,
                            const float* __restrict__ beta,
                            float* __restrict__ tok_row,
                            float* __restrict__ out_ctx) {
  const int b = blockIdx.x, t = threadIdx.x;
  __shared__ float sa[1024], sv[1024], cres[1024], red[256];
  for (int d = t; d < DMODEL; d += 256) {
    sa[d] = cls_attn[b * DMODEL + d];
    sv[d] = qkv0[b * 3072 + 2048 + d];      // v_cls
  }
  __syncthreads();
  for (int j = t; j < DMODEL; j += 256) {
    const float* wr = Wo + (size_t)j * DMODEL;
    float s1 = 0.f, s2 = 0.f;
    for (int d = 0; d < DMODEL; ++d) {
      float w = wr[d];
      s1 += w * sa[d];
      s2 += w * sv[d];
    }
    cres[j] = s1 + bo[j] + ctx[b * DMODEL + j];
    tok_row[b * DMODEL + j] = s2 + bo[j];
  }
  __syncthreads();
  float sum = 0.f, sq = 0.f;
  for (int j = t; j < DMODEL; j += 256) { float v = cres[j]; sum += v; sq += v * v; }
  red[t] = sum; __syncthreads();
  for (int o = 128; o > 0; o >>= 1) { if (t < o) red[t] += red[t + o]; __syncthreads(); }
  float mu = red[0] / (float)DMODEL; __syncthreads();
  red[t] = sq; __syncthreads();
  for (int o = 128; o > 0; o >>= 1) { if (t < o) red[t] += red[t + o]; __syncthreads(); }
  float var = red[0] / (float)DMODEL - mu * mu;
  float rs  = rsqrtf(var + 1e-5f);
  for (int j = t; j < DMODEL; j += 256)
    out_ctx[b * DMODEL + j] = (cres[j] - mu) * rs * gamma[j] + beta[j];
}

// =======================================================================
// Kernel 9: fused token path: res = x + peT + tok_row ; LayerNorm ; write
// feature map in x's own (b,d,hw) layout (reference transpose is free).
// =======================================================================
__global__ void final_kernel(const float* __restrict__ x,
                             const float* __restrict__ peT,
                             const float* __restrict__ tok_row,
                             const float* __restrict__ gamma,
                             const float* __restrict__ beta,
                             float* __restrict__ out) {
  const int b = blockIdx.y, pc = blockIdx.x;   // 256 chunks of 16 pos
  const int p0 = pc * 16, t = threadIdx.x;
  extern __shared__ float tile[];              // [1024][17]
  __shared__ float psum[16][16], psq[16][16], mu_s[16], rs_s[16];
  const size_t xb = (size_t)b * DMODEL * HW;
  for (int i = 0; i < 64; ++i) {
    int e = i * 256 + t;                       // 16384 = 1024*16
    int pl = e & 15, d = e >> 4;
    size_t gi = (size_t)d * HW + p0 + pl;
    tile[d * 17 + pl] = x[xb + gi] + peT[gi] + tok_row[b * DMODEL + d];
  }
  __syncthreads();
  const int pl = t & 15, dg = t >> 4;          // 16 groups of 64 d
  float s = 0.f, q = 0.f;
  for (int d = dg * 64; d < dg * 64 + 64; ++d) {
    float v = tile[d * 17 + pl];
    s += v; q += v * v;
  }
  psum[dg][pl] = s; psq[dg][pl] = q;
  __syncthreads();
  if (t < 16) {
    float ss = 0.f, qq = 0.f;
    for (int g = 0; g < 16; ++g) { ss += psum[g][t]; qq += psq[g][t]; }
    float mu = ss / (float)DMODEL;
    float var = qq / (float)DMODEL - mu * mu;
    mu_s[t] = mu; rs_s[t] = rsqrtf(var + 1e-5f);
  }
  __syncthreads();
  for (int i = 0; i < 64; ++i) {
    int e = i * 256 + t;
    int pl2 = e & 15, d = e >> 4;
    float v = (tile[d * 17 + pl2] - mu_s[pl2]) * rs_s[pl2] * gamma[d] + beta[d];
    out[xb + (size_t)d * HW + p0 + pl2] = v;
  }
}

// =======================================================================
extern "C" void kernel_launch(void* const* d_in, const int* in_sizes, int n_in,
                              void* d_out, int out_size, void* d_ws, size_t ws_size,
                              hipStream_t stream) {
  const float* x     = (const float*)d_in[0];
  const float* ctx   = (const float*)d_in[1];
  const float* Wqkv  = (const float*)d_in[2];
  const float* bqkv  = (const float*)d_in[3];
  const float* Wo    = (const float*)d_in[4];
  const float* bo    = (const float*)d_in[5];
  const float* gamma = (const float*)d_in[6];
  const float* beta  = (const float*)d_in[7];
  float* out     = (float*)d_out;
  float* out_ctx = out + (size_t)BATCH * DMODEL * HW;  // after feature map
  float* ws      = (float*)d_ws;

  float* peT      = ws + OFF_PE;
  float* qkv0     = ws + OFF_QKV0;
  float* u        = ws + OFF_U;
  float* qbk      = ws + OFF_QBK;
  float* scores   = ws + OFF_SC;
  float* m_part   = ws + OFF_MP;
  float* cls_attn = ws + OFF_CA;
  float* tok_row  = ws + OFF_TR;

  pe_kernel<<<(DMODEL * HW) / 256, 256, 0, stream>>>(peT);
  qkv0_wmma_kernel<<<3072 / 16, 32, 0, stream>>>(ctx, Wqkv, bqkv, qkv0);
  u_kernel<<<BATCH * NHEAD, 256, 0, stream>>>(qkv0, Wqkv, bqkv, u, qbk);
  scores_wmma_kernel<<<dim3(32, BATCH), 256, 1024 * 16 * sizeof(float), stream>>>(
      x, peT, u, qbk, scores);
  softmax_kernel<<<BATCH * NHEAD, 256, 0, stream>>>(qkv0, scores);
  maccum_kernel<<<dim3(8, 8, BATCH), 256,
                  (16 * 513 + 128 * 65) * sizeof(float), stream>>>(
      x, peT, scores, m_part);
  clsA_kernel<<<BATCH, 256, 16 * 1024 * sizeof(float), stream>>>(
      m_part, scores, ctx, Wqkv, bqkv, cls_attn);
  clsB_kernel<<<BATCH, 256, 0, stream>>>(cls_attn, qkv0, ctx, Wo, bo, gamma,
                                         beta, tok_row, out_ctx);
  final_kernel<<<dim3(256, BATCH), 256, 1024 * 17 * sizeof(float), stream>>>(
      x, peT, tok_row, gamma, beta, out);
}